// GraphEncoder_72773925863651
// MI455X (gfx1250) — compile-verified
//
#include <hip/hip_runtime.h>
#include <math.h>

// ---------------- problem constants ----------------
#define N_NODES 100
#define NP      112        // padded to 7*16
#define F_IN    27
#define N_EDGE  262144
#define HS      256        // row stride of H / XW buffers (floats)
#define AS      113        // row stride of adjacency (coprime with 64 -> no LDS bank clash)
#define NTHREADS 512       // 16 waves of 32

// workspace layout (float offsets)
#define OFF_DEG  0
#define OFF_DINV 128
#define OFF_A    256
#define OFF_H    12928                    // 256 + 112*113 = 12912, rounded up
#define OFF_XW   (OFF_H + NP*HS)          // 41600
#define OFF_W1P  (OFF_XW + NP*HS)         // 70272 : W1 zero-padded to 28x64
#define OFF_END  (OFF_W1P + 28*64)        // 72064 floats = ~288 KB of d_ws

typedef float v2f __attribute__((ext_vector_type(2)));
typedef float v8f __attribute__((ext_vector_type(8)));

__device__ __forceinline__ v8f wmma_f32_4(v2f a, v2f b, v8f c) {
  // V_WMMA_F32_16X16X4_F32 : D(16x16 f32) = A(16x4 f32) * B(4x16 f32) + C
  return __builtin_amdgcn_wmma_f32_16x16x4_f32(
      /*neg_a=*/false, a, /*neg_b=*/false, b,
      /*c_mod=*/(short)0, c, /*reuse_a=*/false, /*reuse_b=*/false);
}

// ---------------- phase kernels (edge scatter) ----------------

__global__ void k_zero(float* __restrict__ ws, int n) {
  int i = blockIdx.x * blockDim.x + threadIdx.x;
  int stride = gridDim.x * blockDim.x;
  for (; i < n; i += stride) ws[i] = 0.0f;
}

// degree accumulation: LDS-staged histogram (100 hot addresses), then global atomics
__global__ void k_deg(const int* __restrict__ dst, const float* __restrict__ ew,
                      float* __restrict__ gDeg) {
  __shared__ float sdeg[NP];
  int tid = threadIdx.x;
  if (tid < NP) sdeg[tid] = 0.0f;
  __syncthreads();
  int i = blockIdx.x * blockDim.x + tid;
  int stride = gridDim.x * blockDim.x;
  for (; i < N_EDGE; i += stride) atomicAdd(&sdeg[dst[i]], ew[i]);
  __syncthreads();
  if (tid < NP && sdeg[tid] != 0.0f) atomicAdd(&gDeg[tid], sdeg[tid]);
}

// dinv = (deg+1)^-1/2 ; scatter x into padded H ; copy W1 into K-padded buffer
__global__ void k_prep(const float* __restrict__ x, const float* __restrict__ W1,
                       float* __restrict__ ws) {
  float* gDeg  = ws + OFF_DEG;
  float* gDinv = ws + OFF_DINV;
  float* gH    = ws + OFF_H;
  float* gW1   = ws + OFF_W1P;
  int tid = threadIdx.x;
  if (tid < NP) gDinv[tid] = (tid < N_NODES) ? rsqrtf(gDeg[tid] + 1.0f) : 0.0f;
  for (int idx = tid; idx < N_NODES * F_IN; idx += blockDim.x)
    gH[(idx / F_IN) * HS + (idx % F_IN)] = x[idx];
  // rows 0..26 are a straight copy; row 27 stays zero (k_zero)
  for (int idx = tid; idx < F_IN * 64; idx += blockDim.x)
    gW1[idx] = W1[idx];
}

// dense normalized adjacency:  A[d][s] += dinv[s]*ew*dinv[d] ; diag += dinv^2
__global__ void k_adj(const int* __restrict__ src, const int* __restrict__ dst,
                      const float* __restrict__ ew, float* __restrict__ ws) {
  const float* gDinv = ws + OFF_DINV;
  float* gA = ws + OFF_A;
  int tid = threadIdx.x;
  int i = blockIdx.x * blockDim.x + tid;
  int stride = gridDim.x * blockDim.x;
  for (; i < N_EDGE; i += stride) {
    int s = src[i], d = dst[i];
    atomicAdd(&gA[d * AS + s], gDinv[s] * ew[i] * gDinv[d]);
  }
  if (blockIdx.x == 0 && tid < N_NODES)
    atomicAdd(&gA[tid * AS + tid], gDinv[tid] * gDinv[tid]);
}

// ---------------- templated WMMA GEMMs (all dims compile-time) ----------------

// XW = H @ W ; KP = K padded to multiple of 4 (W buffer must have KP zero-padded rows)
template <int KP, int OD>
__device__ __forceinline__ void gemm_xw(const float* __restrict__ gH,
                                        const float* __restrict__ W,
                                        float* __restrict__ gXW,
                                        int wave, int laneN, int kHi, int rHi) {
  constexpr int NTN = OD / 16;          // power of two -> t/NTN is a shift
  constexpr int NT  = 7 * NTN;
  for (int t = wave; t < NT; t += 16) {
    int r0 = (t / NTN) * 16, c0 = (t % NTN) * 16;
    v8f acc = {};
    #pragma unroll 4
    for (int k = 0; k < KP; k += 4) {
      int ak = k + kHi;
      v2f a;
      a.x = gH[(r0 + laneN) * HS + ak];      // padded cols of H are zero
      a.y = gH[(r0 + laneN) * HS + ak + 1];
      v2f b;
      b.x = W[ak * OD + c0 + laneN];         // W pre-padded: no guards
      b.y = W[(ak + 1) * OD + c0 + laneN];
      acc = wmma_f32_4(a, b, acc);
    }
    int rs = r0 + rHi;
    #pragma unroll
    for (int v = 0; v < 8; ++v)
      gXW[(rs + v) * HS + c0 + laneN] = acc[v];
  }
}

// H = relu(Â @ XW + bias) ; A operand streamed from LDS, K = NP = 112
template <int OD>
__device__ __forceinline__ void gemm_agg(const float* __restrict__ sA,
                                         const float* __restrict__ gXW,
                                         const float* __restrict__ bias,
                                         float* __restrict__ gH,
                                         int wave, int laneN, int kHi, int rHi) {
  constexpr int NTN = OD / 16;
  constexpr int NT  = 7 * NTN;
  for (int t = wave; t < NT; t += 16) {
    int r0 = (t / NTN) * 16, c0 = (t % NTN) * 16;
    v8f acc = {};
    #pragma unroll 4
    for (int k = 0; k < NP; k += 4) {
      int ak = k + kHi;
      v2f a;
      a.x = sA[(r0 + laneN) * AS + ak];
      a.y = sA[(r0 + laneN) * AS + ak + 1];
      v2f b;
      b.x = gXW[ak * HS + c0 + laneN];
      b.y = gXW[(ak + 1) * HS + c0 + laneN];
      acc = wmma_f32_4(a, b, acc);
    }
    float bb = bias[c0 + laneN];
    int rs = r0 + rHi;
    #pragma unroll
    for (int v = 0; v < 8; ++v)
      gH[(rs + v) * HS + c0 + laneN] = fmaxf(acc[v] + bb, 0.0f);
  }
}

// ---------------- main persistent kernel ----------------

__global__ __launch_bounds__(NTHREADS) void k_main(
    const float* __restrict__ W2, const float* __restrict__ b1,
    const float* __restrict__ b2,
    const float* __restrict__ W3, const float* __restrict__ b3,
    const float* __restrict__ gamma, const float* __restrict__ beta,
    const float* __restrict__ lw1, const float* __restrict__ lb1,
    const float* __restrict__ lw2, const float* __restrict__ lb2,
    float* __restrict__ ws, float* __restrict__ out) {
  __shared__ float sA[NP * AS];      // ~50.6 KB: dense adjacency (WMMA A operand)
  __shared__ float sVec[1024];       // relu(linear1) result [1000]
  __shared__ float sScale[256];
  __shared__ float sShift[256];

  float* gA  = ws + OFF_A;
  float* gH  = ws + OFF_H;
  float* gXW = ws + OFF_XW;
  const float* gW1 = ws + OFF_W1P;

  const int tid   = threadIdx.x;
  const int lane  = tid & 31;
  const int wave  = tid >> 5;
  const int laneN = lane & 15;
  const int kHi   = (lane >= 16) ? 2 : 0;   // hi half-lanes hold K+2..K+3 (ISA 7.12.2)
  const int rHi   = (lane >= 16) ? 8 : 0;   // C/D: hi half-lanes hold rows M+8..M+15

  // stage adjacency into LDS
  for (int i = tid; i < NP * AS; i += NTHREADS) sA[i] = gA[i];
  __syncthreads();

  // ---- layer 1: 27(->28 padded) -> 64 ----
  gemm_xw<28, 64>(gH, gW1, gXW, wave, laneN, kHi, rHi);
  __threadfence(); __syncthreads();
  gemm_agg<64>(sA, gXW, b1, gH, wave, laneN, kHi, rHi);
  __threadfence(); __syncthreads();

  // ---- layer 2: 64 -> 128 ----
  gemm_xw<64, 128>(gH, W2, gXW, wave, laneN, kHi, rHi);
  __threadfence(); __syncthreads();
  gemm_agg<128>(sA, gXW, b2, gH, wave, laneN, kHi, rHi);
  __threadfence(); __syncthreads();

  // ---- layer 3: 128 -> 256 ----
  gemm_xw<128, 256>(gH, W3, gXW, wave, laneN, kHi, rHi);
  __threadfence(); __syncthreads();
  gemm_agg<256>(sA, gXW, b3, gH, wave, laneN, kHi, rHi);
  __threadfence(); __syncthreads();

  // ---- BatchNorm (batch stats over 100 rows, biased var), folded into scale/shift ----
  for (int c = tid; c < 256; c += NTHREADS) {
    float s = 0.0f, ss = 0.0f;
    for (int i = 0; i < N_NODES; ++i) {
      float v = gH[i * HS + c];
      s += v; ss += v * v;
    }
    float mu  = s * 0.01f;
    float var = ss * 0.01f - mu * mu;
    float sc  = gamma[c] * rsqrtf(var + 1e-5f);
    sScale[c] = sc;
    sShift[c] = beta[c] - mu * sc;
  }
  __syncthreads();

  // ---- linear1 (256 -> 10 per node) + relu -> sVec[1000]  (scatter_max = identity) ----
  for (int idx = tid; idx < N_NODES * 10; idx += NTHREADS) {
    int i = idx / 10, j = idx % 10;
    const float* hr = gH + i * HS;
    const float* wr = lw1 + j * 256;
    float s = lb1[j];
    for (int c = 0; c < 256; ++c)
      s += (hr[c] * sScale[c] + sShift[c]) * wr[c];
    sVec[idx] = fmaxf(s, 0.0f);
  }
  __syncthreads();

  // ---- linear2 (1000 -> 128): one wave per 8 outputs, lane-parallel dot + xor-reduce ----
  for (int q = 0; q < 8; ++q) {
    int m = wave * 8 + q;                 // 16 waves * 8 = 128 outputs
    const float* wr = lw2 + m * 1000;
    float p = 0.0f;
    for (int k = lane; k < 1000; k += 32) p += sVec[k] * wr[k];
    #pragma unroll
    for (int off = 16; off > 0; off >>= 1) p += __shfl_xor(p, off, 32);
    if (lane == 0) out[m] = p + lb2[m];
  }
}

// ---------------- host launch ----------------

extern "C" void kernel_launch(void* const* d_in, const int* in_sizes, int n_in,
                              void* d_out, int out_size, void* d_ws, size_t ws_size,
                              hipStream_t stream) {
  const float* x     = (const float*)d_in[0];
  const int*   ei    = (const int*)  d_in[1];   // [2, E]
  const float* ea    = (const float*)d_in[2];   // [E]
  const float* W1    = (const float*)d_in[3];
  const float* b1    = (const float*)d_in[4];
  const float* W2    = (const float*)d_in[5];
  const float* b2    = (const float*)d_in[6];
  const float* W3    = (const float*)d_in[7];
  const float* b3    = (const float*)d_in[8];
  const float* gamma = (const float*)d_in[9];
  const float* beta  = (const float*)d_in[10];
  const float* lw1   = (const float*)d_in[11];
  const float* lb1   = (const float*)d_in[12];
  const float* lw2   = (const float*)d_in[13];
  const float* lb2   = (const float*)d_in[14];

  float* ws  = (float*)d_ws;
  float* out = (float*)d_out;
  const int* src = ei;
  const int* dst = ei + N_EDGE;

  k_zero<<<256, 256, 0, stream>>>(ws, OFF_END);
  k_deg <<<256, 256, 0, stream>>>(dst, ea, ws + OFF_DEG);
  k_prep<<<1, 256, 0, stream>>>(x, W1, ws);
  k_adj <<<256, 256, 0, stream>>>(src, dst, ea, ws);
  k_main<<<1, NTHREADS, 0, stream>>>(W2, b1, b2, W3, b3, gamma, beta,
                                     lw1, lb1, lw2, lb2, ws, out);
}